// Quantizer_86088324481611
// MI455X (gfx1250) — compile-verified
//
#include <hip/hip_runtime.h>

// VQ-VAE quantizer for MI455X (gfx1250).
//   inputs : z_e [1024,64,16,16] f32, e [512,64] f32, N [512] f32, m [512,64] f32
//   outputs: z_q [1024,64,16,16] | e_new [512,64] | N_new [512] | m_new [512,64]
//
// Phase plan (all on `stream`):
//   1) zero 133 KB of workspace (cnt[512], msum[512*64])
//   2) vq_quantize: fp32 WMMA distance GEMM + argmax + z_q write + scatter
//   3) vq_finalize: EMA update of N / m / e

#define K_CODES   512
#define DIM       64
#define HWB       256                // H*W per batch image
#define PITCH     513                // LDS pitch: 513 % 64 == 1 -> conflict-free
#define NEG_INF   (-3.4e38f)

typedef float v2f __attribute__((ext_vector_type(2)));
typedef float v8f __attribute__((ext_vector_type(8)));

__global__ void vq_zero_ws(float* __restrict__ ws, int n) {
    int i = blockIdx.x * blockDim.x + threadIdx.x;
    if (i < n) ws[i] = 0.0f;
}

__global__ __launch_bounds__(256)
void vq_quantize(const float* __restrict__ z_e,
                 const float* __restrict__ e,
                 float* __restrict__ z_q,
                 float* __restrict__ cnt_ws,   // [512]
                 float* __restrict__ m_ws) {   // [512*64]
    extern __shared__ float smem[];
    float*    eT      = smem;                          // [64][PITCH]
    float*    nh_e2   = smem + DIM * PITCH;            // [512]  (-0.5*||e||^2)
    unsigned* idx_loc = (unsigned*)(nh_e2 + K_CODES);  // [256]
    unsigned* hist    = idx_loc + HWB;                 // [512]

    const int tid  = threadIdx.x;
    const int wg   = blockIdx.x;        // == batch index b (256 rows per image)
    const int lane = tid & 31;
    const int wave = tid >> 5;
    const int mrow = lane & 15;         // M (row within 16x16 tile) / N for B,C
    const int kh   = lane >> 4;         // K-half select for A/B fragments

    // ---- stage codebook transposed into LDS (coalesced global, padded LDS) ----
    const float4* e4 = (const float4*)e;
    for (int i = tid; i < K_CODES * DIM / 4; i += 256) {
        int code = i >> 4;
        int d0   = (i & 15) << 2;
        float4 v = e4[i];
        eT[(d0 + 0) * PITCH + code] = v.x;
        eT[(d0 + 1) * PITCH + code] = v.y;
        eT[(d0 + 2) * PITCH + code] = v.z;
        eT[(d0 + 3) * PITCH + code] = v.w;
    }
    for (int c = tid; c < K_CODES; c += 256) hist[c] = 0u;
    __syncthreads();

    for (int c = tid; c < K_CODES; c += 256) {
        float s = 0.0f;
        #pragma unroll 8
        for (int d = 0; d < DIM; ++d) { float t = eT[d * PITCH + c]; s += t * t; }
        nh_e2[c] = -0.5f * s;
    }
    __syncthreads();

    // ---- per-wave: 32 rows, score = z.e - 0.5||e||^2, argmax over 512 codes ----
    const int rowloc = wave * 32;                    // local row base (2 tiles of 16)
    const float* z0 = z_e + (size_t)wg * (DIM * HWB) + (rowloc + mrow);
    const float* z1 = z0 + 16;

    // A fragments resident: 2 row-tiles x 16 k-steps x v2f  (16x4 f32 layout:
    // VGPR0 = K{0,2}, VGPR1 = K{1,3}, M = lane%16)
    v2f a0[16], a1[16];
    #pragma unroll
    for (int ks = 0; ks < 16; ++ks) {
        int kA = 4 * ks + 2 * kh;
        a0[ks].x = z0[(kA    ) * HWB];
        a0[ks].y = z0[(kA + 1) * HWB];
        a1[ks].x = z1[(kA    ) * HWB];
        a1[ks].y = z1[(kA + 1) * HWB];
    }

    float bv0[8], bv1[8]; unsigned bi0[8], bi1[8];
    #pragma unroll
    for (int j = 0; j < 8; ++j) {
        bv0[j] = NEG_INF; bv1[j] = NEG_INF; bi0[j] = 0u; bi1[j] = 0u;
    }

    for (int ct = 0; ct < 32; ++ct) {
        int c0 = ct * 16;
        float ci = nh_e2[c0 + mrow];                 // C init per lane: -0.5||e_N||^2
        v8f acc0 = {ci, ci, ci, ci, ci, ci, ci, ci};
        v8f acc1 = acc0;
        #pragma unroll
        for (int ks = 0; ks < 16; ++ks) {
            int kB = 4 * ks + 2 * kh;
            v2f bf;
            bf.x = eT[(kB    ) * PITCH + c0 + mrow];
            bf.y = eT[(kB + 1) * PITCH + c0 + mrow];
            acc0 = __builtin_amdgcn_wmma_f32_16x16x4_f32(
                false, a0[ks], false, bf, (short)0, acc0, false, false);
            acc1 = __builtin_amdgcn_wmma_f32_16x16x4_f32(
                false, a1[ks], false, bf, (short)0, acc1, false, false);
        }
        unsigned cidx = (unsigned)(c0 + mrow);
        #pragma unroll
        for (int j = 0; j < 8; ++j) {
            float v0 = acc0[j];
            if (v0 > bv0[j] || (v0 == bv0[j] && cidx < bi0[j])) { bv0[j] = v0; bi0[j] = cidx; }
            float v1 = acc1[j];
            if (v1 > bv1[j] || (v1 == bv1[j] && cidx < bi1[j])) { bv1[j] = v1; bi1[j] = cidx; }
        }
    }

    // cross-lane (val,idx) max-reduce within each 16-lane half (ties -> lower idx)
    #pragma unroll
    for (int j = 0; j < 8; ++j) {
        #pragma unroll
        for (int off = 1; off < 16; off <<= 1) {
            float    ov = __shfl_xor(bv0[j], off, 32);
            unsigned oi = (unsigned)__shfl_xor((int)bi0[j], off, 32);
            if (ov > bv0[j] || (ov == bv0[j] && oi < bi0[j])) { bv0[j] = ov; bi0[j] = oi; }
            ov = __shfl_xor(bv1[j], off, 32);
            oi = (unsigned)__shfl_xor((int)bi1[j], off, 32);
            if (ov > bv1[j] || (ov == bv1[j] && oi < bi1[j])) { bv1[j] = ov; bi1[j] = oi; }
        }
    }
    if (mrow == 0) {                     // lanes 0 and 16: rows M=0..7 / M=8..15
        int rhalf = kh * 8;
        #pragma unroll
        for (int j = 0; j < 8; ++j) {
            idx_loc[rowloc +  0 + rhalf + j] = bi0[j];
            idx_loc[rowloc + 16 + rhalf + j] = bi1[j];
            atomicAdd(&hist[bi0[j]], 1u);
            atomicAdd(&hist[bi1[j]], 1u);
        }
    }
    __syncthreads();

    // ---- fused z_q gather-write (coalesced) + m_i scatter (L2 fp32 atomics) ----
    const float* zin = z_e + (size_t)wg * (DIM * HWB);
    float*       zq  = z_q + (size_t)wg * (DIM * HWB);
    for (int it = 0; it < 64; ++it) {
        int id = it * 256 + tid;         // id = d*256 + hw, matches [D][HW] layout
        int d  = id >> 8;
        int hw = id & 255;
        unsigned code = idx_loc[hw];
        zq[id] = eT[d * PITCH + code];
        atomicAdd(&m_ws[code * DIM + d], zin[id]);
    }
    for (int c = tid; c < K_CODES; c += 256) {
        unsigned h = hist[c];
        if (h) atomicAdd(&cnt_ws[c], (float)h);
    }
}

__global__ __launch_bounds__(256)
void vq_finalize(const float* __restrict__ N_in, const float* __restrict__ m_in,
                 const float* __restrict__ cnt_ws, const float* __restrict__ m_ws,
                 float* __restrict__ e_out, float* __restrict__ N_out,
                 float* __restrict__ m_out) {
    int i = blockIdx.x * blockDim.x + threadIdx.x;   // 0..32767 (k*64 + d)
    int k = i >> 6;
    float Nn = 0.99f * N_in[k] + 0.01f * cnt_ws[k];
    float mn = 0.99f * m_in[i] + 0.01f * m_ws[i];
    e_out[i] = mn / (Nn + 1e-8f);
    m_out[i] = mn;
    if ((i & 63) == 0) N_out[k] = Nn;
}

extern "C" void kernel_launch(void* const* d_in, const int* in_sizes, int n_in,
                              void* d_out, int out_size, void* d_ws, size_t ws_size,
                              hipStream_t stream) {
    (void)in_sizes; (void)n_in; (void)out_size; (void)ws_size;

    const float* z_e  = (const float*)d_in[0];   // [1024,64,16,16]
    const float* e    = (const float*)d_in[1];   // [512,64]
    const float* N_in = (const float*)d_in[2];   // [512]
    const float* m_in = (const float*)d_in[3];   // [512,64]

    float* out   = (float*)d_out;
    float* z_q   = out;                               // 16,777,216
    float* e_out = out + (size_t)1024 * 64 * 256;     //     32,768
    float* N_out = e_out + 512 * 64;                  //        512
    float* m_out = N_out + 512;                       //     32,768

    float* cnt_ws = (float*)d_ws;                     // [512]
    float* m_ws   = cnt_ws + K_CODES;                 // [512*64]
    const int ws_elems = K_CODES + K_CODES * DIM;     // 33,280 floats (133 KB)

    vq_zero_ws<<<(ws_elems + 255) / 256, 256, 0, stream>>>(cnt_ws, ws_elems);

    const size_t smem_bytes =
        (DIM * PITCH + K_CODES) * sizeof(float) + (HWB + K_CODES) * sizeof(unsigned);
    vq_quantize<<<1024, 256, smem_bytes, stream>>>(z_e, e, z_q, cnt_ws, m_ws);

    vq_finalize<<<(K_CODES * DIM) / 256, 256, 0, stream>>>(
        N_in, m_in, cnt_ws, m_ws, e_out, N_out, m_out);
}